// MoNet_90400471646622
// MI455X (gfx1250) — compile-verified
//
#include <hip/hip_runtime.h>
#include <hip/hip_bf16.h>
#include <math.h>

// ---------------- problem constants (match reference) ----------------
#define NN   50000
#define EE   800000
#define FIN  128
#define HID  64
#define NCLS 16
#define KK   3
#define DD   2
#define EPSV 1e-15f

typedef __attribute__((ext_vector_type(2))) float v2f;
typedef __attribute__((ext_vector_type(8))) float v8f;

// =====================================================================
// fp32 WMMA GEMM: C[M,Ncols] = A[M,Kdim] * B[Kdim,Ncols]
// One wave (32 threads) computes one 16x16 tile via V_WMMA_F32_16X16X4_F32.
// Requires: M % 16 == 0, Ncols % 16 == 0, Kdim % 4 == 0  (all true here).
// Fragment layouts per CDNA5 ISA 7.12.2:
//   A 16x4 : lanes 0-15 rows M=0..15 hold K=0(k VGPR0),1(VGPR1);
//            lanes 16-31 same rows hold K=2,3.
//   B 4x16 : mirrored (cols across lanes, K pair per half-wave).
//   C/D    : VGPR r -> row (half*8 + r), col = lane&15.
// =====================================================================
__global__ __launch_bounds__(32) void gemm_wmma_f32(
    const float* __restrict__ A, const float* __restrict__ B,
    float* __restrict__ C, int Kdim, int Ncols) {
  const int n0   = blockIdx.x * 16;
  const int m0   = blockIdx.y * 16;
  const int lane = threadIdx.x;       // 0..31
  const int half = lane >> 4;         // 0 or 1
  const int l16  = lane & 15;

  v8f acc = {};
  const float* Arow = A + (size_t)(m0 + l16) * Kdim;
  #pragma unroll 4
  for (int k = 0; k < Kdim; k += 4) {
    const int kb = k + half * 2;
    v2f a, b;
    a.x = Arow[kb];
    a.y = Arow[kb + 1];
    const float* Bp = B + (size_t)kb * Ncols + (n0 + l16);
    b.x = Bp[0];
    b.y = Bp[Ncols];
    acc = __builtin_amdgcn_wmma_f32_16x16x4_f32(
        /*neg_a=*/false, a, /*neg_b=*/false, b,
        /*c_mod=*/(short)0, acc, /*reuse_a=*/false, /*reuse_b=*/false);
  }
  float* Cp = C + (size_t)(m0 + half * 8) * Ncols + (n0 + l16);
  #pragma unroll
  for (int r = 0; r < 8; ++r)
    Cp[(size_t)r * Ncols] = acc[r];
}

// =====================================================================
// Edge message pass, layer 1: 64 threads per edge (one per output channel m).
// msg[m] = sum_k gauss(edge_attr, mu, sigma)[k] * H[src, k*64 + m]
// atomicAdd into agg[dst, m]; lane m==0 also counts degree.
// =====================================================================
__global__ __launch_bounds__(256) void edge_msg1(
    const float* __restrict__ H,      // [N, K*HID] = [N,192]
    const int* __restrict__ src, const int* __restrict__ dst,
    const float* __restrict__ ea,     // [E, 2]
    const float* __restrict__ mu,     // [3,2]
    const float* __restrict__ sigma,  // [3,2]
    float* __restrict__ agg,          // [N, 64]
    float* __restrict__ deg) {
  const int t = blockIdx.x * blockDim.x + threadIdx.x;
  const int e = t >> 6;
  const int m = t & 63;
  if (e >= EE) return;
  const int s = src[e];
  const int d = dst[e];
  const float ax = ea[2 * e + 0];
  const float ay = ea[2 * e + 1];
  const float* Hs = H + (size_t)s * (KK * HID);
  float msg = 0.f;
  #pragma unroll
  for (int k = 0; k < KK; ++k) {
    const float dx = ax - mu[2 * k + 0];
    const float dy = ay - mu[2 * k + 1];
    const float sx = sigma[2 * k + 0];
    const float sy = sigma[2 * k + 1];
    const float w = expf(-0.5f * (dx * dx / (EPSV + sx * sx) +
                                  dy * dy / (EPSV + sy * sy)));
    msg += Hs[k * HID + m] * w;
  }
  atomicAdd(&agg[(size_t)d * HID + m], msg);
  if (m == 0) atomicAdd(&deg[d], 1.0f);
}

// Edge message pass, layer 2: 16 threads per edge (NCLS=16).
__global__ __launch_bounds__(256) void edge_msg2(
    const float* __restrict__ H,      // [N, K*NCLS] = [N,48]
    const int* __restrict__ src, const int* __restrict__ dst,
    const float* __restrict__ ea,
    const float* __restrict__ mu,
    const float* __restrict__ sigma,
    float* __restrict__ agg) {        // [N, 16]
  const int t = blockIdx.x * blockDim.x + threadIdx.x;
  const int e = t >> 4;
  const int m = t & 15;
  if (e >= EE) return;
  const int s = src[e];
  const int d = dst[e];
  const float ax = ea[2 * e + 0];
  const float ay = ea[2 * e + 1];
  const float* Hs = H + (size_t)s * (KK * NCLS);
  float msg = 0.f;
  #pragma unroll
  for (int k = 0; k < KK; ++k) {
    const float dx = ax - mu[2 * k + 0];
    const float dy = ay - mu[2 * k + 1];
    const float sx = sigma[2 * k + 0];
    const float sy = sigma[2 * k + 1];
    const float w = expf(-0.5f * (dx * dx / (EPSV + sx * sx) +
                                  dy * dy / (EPSV + sy * sy)));
    msg += Hs[k * NCLS + m] * w;
  }
  atomicAdd(&agg[(size_t)d * NCLS + m], msg);
}

// finalize layer 1: mean-agg + root term + bias, then ELU (alpha=1).
__global__ __launch_bounds__(256) void finalize1(
    const float* __restrict__ agg, const float* __restrict__ deg,
    const float* __restrict__ r1, const float* __restrict__ b1,
    float* __restrict__ hact) {
  const int t = blockIdx.x * blockDim.x + threadIdx.x;
  if (t >= NN * HID) return;
  const int n = t >> 6;
  const int m = t & 63;
  float v = agg[t] / fmaxf(deg[n], 1.0f) + r1[t] + b1[m];
  hact[t] = (v > 0.0f) ? v : expm1f(v);
}

// finalize layer 2 + log_softmax over 16 classes (one thread per node).
__global__ __launch_bounds__(256) void finalize2(
    const float* __restrict__ agg, const float* __restrict__ deg,
    const float* __restrict__ r2, const float* __restrict__ b2,
    float* __restrict__ out) {
  const int n = blockIdx.x * blockDim.x + threadIdx.x;
  if (n >= NN) return;
  const float dg = fmaxf(deg[n], 1.0f);
  float vals[NCLS];
  float mx = -INFINITY;
  #pragma unroll
  for (int m = 0; m < NCLS; ++m) {
    const float v = agg[(size_t)n * NCLS + m] / dg +
                    r2[(size_t)n * NCLS + m] + b2[m];
    vals[m] = v;
    mx = fmaxf(mx, v);
  }
  float s = 0.f;
  #pragma unroll
  for (int m = 0; m < NCLS; ++m) s += expf(vals[m] - mx);
  const float lse = mx + logf(s);
  #pragma unroll
  for (int m = 0; m < NCLS; ++m)
    out[(size_t)n * NCLS + m] = vals[m] - lse;
}

// =====================================================================
extern "C" void kernel_launch(void* const* d_in, const int* in_sizes, int n_in,
                              void* d_out, int out_size, void* d_ws, size_t ws_size,
                              hipStream_t stream) {
  (void)in_sizes; (void)n_in; (void)out_size; (void)ws_size;

  // setup_inputs() order:
  const float* x     = (const float*)d_in[0];   // [N, 128]
  const int*   ei    = (const int*)d_in[1];     // [2, E] (int32 after JAX canonicalization)
  const float* ea    = (const float*)d_in[2];   // [E, 2]
  const float* g1    = (const float*)d_in[3];   // [128, 192]
  const float* mu1   = (const float*)d_in[4];   // [3, 2]
  const float* sig1  = (const float*)d_in[5];   // [3, 2]
  const float* root1 = (const float*)d_in[6];   // [128, 64]
  const float* b1    = (const float*)d_in[7];   // [64]
  const float* g2    = (const float*)d_in[8];   // [64, 48]
  const float* mu2   = (const float*)d_in[9];   // [3, 2]
  const float* sig2  = (const float*)d_in[10];  // [3, 2]
  const float* root2 = (const float*)d_in[11];  // [64, 16]
  const float* b2    = (const float*)d_in[12];  // [16]
  float* out = (float*)d_out;                   // [N, 16]

  const int* src = ei;
  const int* dst = ei + EE;

  // -------- workspace layout (bytes, 256-aligned) --------
  char* ws = (char*)d_ws;
  size_t off = 0;
  auto carve = [&](size_t bytes) {
    char* p = ws + off;
    off += (bytes + 255) & ~(size_t)255;
    return p;
  };
  float* H1   = (float*)carve((size_t)NN * KK * HID * 4);   // 38.4 MB
  float* R1   = (float*)carve((size_t)NN * HID * 4);        // 12.8 MB
  float* AGG1 = (float*)carve((size_t)NN * HID * 4);        // 12.8 MB
  float* DEG  = (float*)carve((size_t)NN * 4);              // 0.2 MB
  float* HACT = (float*)carve((size_t)NN * HID * 4);        // 12.8 MB
  float* H2   = (float*)carve((size_t)NN * KK * NCLS * 4);  //  9.6 MB
  float* R2   = (float*)carve((size_t)NN * NCLS * 4);       //  3.2 MB
  float* AGG2 = (float*)carve((size_t)NN * NCLS * 4);       //  3.2 MB  (~93.0 MB total)

  // zero the atomic accumulators (graph-capture-safe async memsets)
  hipMemsetAsync(AGG1, 0, (size_t)NN * HID * 4, stream);
  hipMemsetAsync(DEG,  0, (size_t)NN * 4, stream);
  hipMemsetAsync(AGG2, 0, (size_t)NN * NCLS * 4, stream);

  const int MT = NN / 16;  // 3125 row tiles

  // ---- layer 1 ----
  gemm_wmma_f32<<<dim3((KK * HID) / 16, MT), 32, 0, stream>>>(x, g1, H1, FIN, KK * HID);
  gemm_wmma_f32<<<dim3(HID / 16, MT), 32, 0, stream>>>(x, root1, R1, FIN, HID);

  {
    const long long threads = (long long)EE * HID;
    edge_msg1<<<(unsigned)((threads + 255) / 256), 256, 0, stream>>>(
        H1, src, dst, ea, mu1, sig1, AGG1, DEG);
  }
  finalize1<<<(NN * HID + 255) / 256, 256, 0, stream>>>(AGG1, DEG, R1, b1, HACT);

  // ---- layer 2 ----
  gemm_wmma_f32<<<dim3((KK * NCLS) / 16, MT), 32, 0, stream>>>(HACT, g2, H2, HID, KK * NCLS);
  gemm_wmma_f32<<<dim3(NCLS / 16, MT), 32, 0, stream>>>(HACT, root2, R2, HID, NCLS);

  {
    const long long threads = (long long)EE * NCLS;
    edge_msg2<<<(unsigned)((threads + 255) / 256), 256, 0, stream>>>(
        H2, src, dst, ea, mu2, sig2, AGG2);
  }
  finalize2<<<(NN + 255) / 256, 256, 0, stream>>>(AGG2, DEG, R2, b2, out);
}